// GraphBlock_35708358099381
// MI455X (gfx1250) — compile-verified
//
#include <hip/hip_runtime.h>

typedef __attribute__((ext_vector_type(2))) float v2f;
typedef __attribute__((ext_vector_type(8))) float v8f;

#define WF      128
#define NGRAPH  64
#define EPSF    1e-5f
#define NPB     128   // nodes per block in segment-reduction kernels

// ---------------- utility: zero a float buffer (graph-capture safe) ----------
__global__ void k_zero(float* __restrict__ p, int n) {
    int i = blockIdx.x * blockDim.x + threadIdx.x;
    int stride = gridDim.x * blockDim.x;
    for (; i < n; i += stride) p[i] = 0.f;
}

// ---------------- GraphNorm pass 1: per-graph sum + count --------------------
// blockDim = 128 (one thread per feature). Each block walks NPB consecutive
// nodes; batch_map is sorted, so we accumulate in registers and flush to
// global atomics only on graph-id change (few flushes per block).
__global__ void k_stats(const float* __restrict__ x, const int* __restrict__ batch,
                        float* __restrict__ gsum, float* __restrict__ gcnt, int n) {
    const int t = threadIdx.x;
    int start = blockIdx.x * NPB;
    if (start >= n) return;
    int end = start + NPB; if (end > n) end = n;
    int cur = batch[start];
    float acc = 0.f, cnt = 0.f;
    for (int i = start; i < end; ++i) {
        int g = batch[i];
        if (g != cur) {
            atomicAdd(&gsum[cur * WF + t], acc);
            if (t == 0) atomicAdd(&gcnt[cur], cnt);
            acc = 0.f; cnt = 0.f; cur = g;
        }
        acc += x[(long)i * WF + t];
        cnt += 1.f;
    }
    atomicAdd(&gsum[cur * WF + t], acc);
    if (t == 0) atomicAdd(&gcnt[cur], cnt);
}

// mean[g][w] = sum/cnt   (in place on gsum)
__global__ void k_mean(float* __restrict__ gsum, const float* __restrict__ gcnt) {
    int i = blockIdx.x * blockDim.x + threadIdx.x;   // < NGRAPH*WF
    float c = gcnt[i >> 7];
    gsum[i] = gsum[i] / (c > 0.f ? c : 1.f);
}

// pass 2: xx = x - mean_scale*mean[batch]; accumulate per-graph sum(out^2)
__global__ void k_center_var(const float* __restrict__ x, const int* __restrict__ batch,
                             const float* __restrict__ mean, const float* __restrict__ mscale,
                             float* __restrict__ xx, float* __restrict__ gvar, int n) {
    const int t = threadIdx.x;
    int start = blockIdx.x * NPB;
    if (start >= n) return;
    int end = start + NPB; if (end > n) end = n;
    const float ms = mscale[t];
    int cur = batch[start];
    float mu = ms * mean[cur * WF + t];
    float vacc = 0.f;
    for (int i = start; i < end; ++i) {
        int g = batch[i];
        if (g != cur) {
            atomicAdd(&gvar[cur * WF + t], vacc);
            vacc = 0.f; cur = g;
            mu = ms * mean[cur * WF + t];
        }
        float o = x[(long)i * WF + t] - mu;
        xx[(long)i * WF + t] = o;
        vacc += o * o;
    }
    atomicAdd(&gvar[cur * WF + t], vacc);
}

// inv_std[g][w] = rsqrt(var_sum/cnt + eps)   (in place on gvar)
__global__ void k_invstd(float* __restrict__ gvar, const float* __restrict__ gcnt) {
    int i = blockIdx.x * blockDim.x + threadIdx.x;   // < NGRAPH*WF
    float c = gcnt[i >> 7];
    c = c > 0.f ? c : 1.f;
    gvar[i] = rsqrtf(gvar[i] / c + EPSF);
}

// xx = weight * xx * inv_std[batch] + bias   (element-wise, in place)
__global__ void k_norm(float* __restrict__ xx, const int* __restrict__ batch,
                       const float* __restrict__ invstd, const float* __restrict__ w,
                       const float* __restrict__ b) {
    int i = blockIdx.x * blockDim.x + threadIdx.x;   // over N*WF
    int t = i & (WF - 1);
    int node = i >> 7;
    int g = batch[node];
    xx[i] = w[t] * xx[i] * invstd[g * WF + t] + b[t];
}

// ---------------- edge scatter: agg[dst] += ew * xin[src] --------------------
// One wave32 per edge: lane l covers features 4l..4l+3 (float4 gather,
// four global_atomic_add_f32 per lane). Edge scalars are wave-uniform.
__global__ void k_scatter(const float* __restrict__ xin, const int* __restrict__ src,
                          const int* __restrict__ dst, const float* __restrict__ ew,
                          float* __restrict__ agg, int e_cnt) {
    int gid = blockIdx.x * blockDim.x + threadIdx.x;
    int e = gid >> 5;
    int lane = threadIdx.x & 31;
    if (e >= e_cnt) return;
    int s = src[e], d = dst[e];
    float w = ew[e];
    const float4 v = ((const float4*)(xin + (long)s * WF))[lane];
    float* ap = agg + (long)d * WF + lane * 4;
    atomicAdd(ap + 0, v.x * w);
    atomicAdd(ap + 1, v.y * w);
    atomicAdd(ap + 2, v.z * w);
    atomicAdd(ap + 3, v.w * w);
}

// ---------------- fused GEMM: out = f(agg*Wrel^T + brel + xin*Wroot^T) -------
// One wave per 16x16 output tile; block of 256 = 8 waves = one 16-row stripe.
// FP32 WMMA 16x16x4: A frag = float2 (lane%16 = M row, (lane/16)*2 = K pair),
// B frag = float2 with B[k][n] = W[n][k] (lane%16 = N col -> contiguous load
// from the row-major weight row). C/D: VGPR j -> M = j + 8*(lane/16), N = lane%16.
__global__ void k_gemm(const float* __restrict__ agg,  const float* __restrict__ Wrel,
                       const float* __restrict__ brel, const float* __restrict__ xin,
                       const float* __restrict__ Wroot, const float* __restrict__ xres,
                       float* __restrict__ out, int apply_leaky) {
    const int lane = threadIdx.x & 31;
    const int tile_n = threadIdx.x >> 5;      // 0..7 (column tile)
    const int tile_m = blockIdx.x;            // row tile
    const int lm   = lane & 15;
    const int koff = (lane >> 4) << 1;        // 0 or 2

    const float* aRow  = agg   + (long)(tile_m * 16 + lm) * WF + koff;
    const float* xRow  = xin   + (long)(tile_m * 16 + lm) * WF + koff;
    const float* bRel  = Wrel  + (long)(tile_n * 16 + lm) * WF + koff;
    const float* bRoot = Wroot + (long)(tile_n * 16 + lm) * WF + koff;

    v8f c = {0.f, 0.f, 0.f, 0.f, 0.f, 0.f, 0.f, 0.f};
#pragma unroll 8
    for (int k0 = 0; k0 < WF; k0 += 4) {
        v2f a0 = *(const v2f*)(aRow + k0);
        v2f b0 = *(const v2f*)(bRel + k0);
        c = __builtin_amdgcn_wmma_f32_16x16x4_f32(false, a0, false, b0,
                                                  (short)0, c, false, false);
        v2f a1 = *(const v2f*)(xRow + k0);
        v2f b1 = *(const v2f*)(bRoot + k0);
        c = __builtin_amdgcn_wmma_f32_16x16x4_f32(false, a1, false, b1,
                                                  (short)0, c, false, false);
    }

    const int col  = tile_n * 16 + lm;
    const float bias = brel[col];
    const int rowBase = tile_m * 16 + ((lane >> 4) << 3);
#pragma unroll
    for (int j = 0; j < 8; ++j) {
        float v = c[j] + bias;
        long oidx = (long)(rowBase + j) * WF + col;
        if (apply_leaky) {
            v = v >= 0.f ? v : 0.1f * v;       // LeakyReLU(0.1)
        } else {
            v += xres[oidx];                    // residual x + conv2(...)
        }
        out[oidx] = v;
    }
}

extern "C" void kernel_launch(void* const* d_in, const int* in_sizes, int n_in,
                              void* d_out, int out_size, void* d_ws, size_t ws_size,
                              hipStream_t stream) {
    (void)n_in; (void)out_size; (void)ws_size;
    const float* x      = (const float*)d_in[0];
    const int*   eidx   = (const int*)  d_in[1];
    const float* ew     = (const float*)d_in[2];
    const int*   batch  = (const int*)  d_in[3];
    const float* gnw    = (const float*)d_in[4];
    const float* gnb    = (const float*)d_in[5];
    const float* gnms   = (const float*)d_in[6];
    const float* Wrel1  = (const float*)d_in[7];
    const float* brel1  = (const float*)d_in[8];
    const float* Wroot1 = (const float*)d_in[9];
    const float* Wrel2  = (const float*)d_in[10];
    const float* brel2  = (const float*)d_in[11];
    const float* Wroot2 = (const float*)d_in[12];
    float* out = (float*)d_out;

    const int N = in_sizes[0] / WF;     // 100000
    const int E = in_sizes[2];          // 625000
    const int* src = eidx;
    const int* dst = eidx + E;

    float* ws = (float*)d_ws;
    size_t off = 0;
    float* xx   = ws + off; off += (size_t)N * WF;
    float* y    = ws + off; off += (size_t)N * WF;
    float* agg  = ws + off; off += (size_t)N * WF;
    float* gsum = ws + off; off += (size_t)NGRAPH * WF;   // becomes mean
    float* gvar = ws + off; off += (size_t)NGRAPH * WF;   // becomes inv_std
    float* gcnt = ws + off; off += (size_t)NGRAPH;

    const int statBlocks = (N + NPB - 1) / NPB;
    const int nwTotal = N * WF;                  // 12,800,000
    const int scatBlocks = (E * 32 + 255) / 256; // 1 wave per edge

    // GraphNorm
    k_zero<<<64, 256, 0, stream>>>(gsum, 2 * NGRAPH * WF + NGRAPH);
    k_stats<<<statBlocks, WF, 0, stream>>>(x, batch, gsum, gcnt, N);
    k_mean<<<(NGRAPH * WF) / 256, 256, 0, stream>>>(gsum, gcnt);
    k_center_var<<<statBlocks, WF, 0, stream>>>(x, batch, gsum, gnms, xx, gvar, N);
    k_invstd<<<(NGRAPH * WF) / 256, 256, 0, stream>>>(gvar, gcnt);
    k_norm<<<nwTotal / 256, 256, 0, stream>>>(xx, batch, gvar, gnw, gnb);

    // Conv1 + LeakyReLU
    k_zero<<<4096, 256, 0, stream>>>(agg, nwTotal);
    k_scatter<<<scatBlocks, 256, 0, stream>>>(xx, src, dst, ew, agg, E);
    k_gemm<<<N / 16, 256, 0, stream>>>(agg, Wrel1, brel1, xx, Wroot1, x, y, 1);

    // Conv2 + residual
    k_zero<<<4096, 256, 0, stream>>>(agg, nwTotal);
    k_scatter<<<scatBlocks, 256, 0, stream>>>(y, src, dst, ew, agg, E);
    k_gemm<<<N / 16, 256, 0, stream>>>(agg, Wrel2, brel2, y, Wroot2, x, out, 0);
}